// CNNMAD_22342419874068
// MI455X (gfx1250) — compile-verified
//
#include <hip/hip_runtime.h>

#define N_  1024
#define NY_ 1024
#define T_  256
#define TP_ 256
#define D_  8
#define C_  8
#define K_  (TP_*D_)   // 2048

typedef __attribute__((ext_vector_type(2))) float v2f;
typedef __attribute__((ext_vector_type(8))) float v8f;

// ---------------- Kernel 1: row/col sums of the DTW path matrices ----------
__global__ void k_sums(const float* __restrict__ pi,
                       float* __restrict__ rowsum, float* __restrict__ colsum) {
    int c = blockIdx.x;          // class
    int t = threadIdx.x;         // index into T (==TP)
    const float* P = pi + (size_t)c * T_ * TP_;
    float rs = 0.f, cs = 0.f;
    for (int j = 0; j < TP_; ++j) rs += P[t * TP_ + j];
    for (int i = 0; i < T_;  ++i) cs += P[i * TP_ + t];
    rowsum[c * T_  + t] = rs;
    colsum[c * TP_ + t] = cs;
}

// ---------------- Kernel 2: C2[c][j] = sum_tp |Y[j,tp]|^2 * colsum[c,tp] ----
__global__ void k_c2(const float* __restrict__ Y,
                     const float* __restrict__ colsum, float* __restrict__ C2) {
    __shared__ float y2[TP_];
    __shared__ float part[C_][32];
    int j   = blockIdx.x;
    int tid = threadIdx.x;       // 256 threads = 8 waves
    const float4* Yv = (const float4*)(Y + (size_t)j * TP_ * D_);
    float4 a = Yv[tid * 2 + 0];
    float4 b = Yv[tid * 2 + 1];
    y2[tid] = a.x*a.x + a.y*a.y + a.z*a.z + a.w*a.w
            + b.x*b.x + b.y*b.y + b.z*b.z + b.w*b.w;
    __syncthreads();
    int w = tid >> 5;            // wave id == class id
    int lane = tid & 31;
    float p = 0.f;
    for (int tp = lane; tp < TP_; tp += 32) p += y2[tp] * colsum[w * TP_ + tp];
    part[w][lane] = p;
    __syncthreads();
    if (tid < C_) {
        float s = 0.f;
        for (int l = 0; l < 32; ++l) s += part[tid][l];
        C2[tid * NY_ + j] = s;
    }
}

// ---------------- Kernel 3: C1n[n] = sum_t |X[n,t]|^2 * rowsum[class,t] ----
__global__ void k_c1(const float* __restrict__ X, const float* __restrict__ rowsum,
                     const int* __restrict__ classes, float* __restrict__ C1n) {
    __shared__ float red[256];
    int n = blockIdx.x, t = threadIdx.x;
    int c = classes[n];
    const float4* Xv = (const float4*)(X + (size_t)n * T_ * D_);
    float4 a = Xv[t * 2 + 0], b = Xv[t * 2 + 1];
    float x2 = a.x*a.x + a.y*a.y + a.z*a.z + a.w*a.w
             + b.x*b.x + b.y*b.y + b.z*b.z + b.w*b.w;
    red[t] = x2 * rowsum[c * T_ + t];
    __syncthreads();
    for (int s = 128; s > 0; s >>= 1) {
        if (t < s) red[t] += red[t + s];
        __syncthreads();
    }
    if (t == 0) C1n[n] = red[0];
}

// ---------------- Kernel 4: XW[n,tp,:] = sum_t pi[c,t,tp] * X[n,t,:] -------
__global__ void k_warp(const float* __restrict__ X, const float* __restrict__ pi,
                       const int* __restrict__ classes, float* __restrict__ XW) {
    __shared__ float Xl[T_ * D_];           // 8 KB: full sample
    int n = blockIdx.x, tid = threadIdx.x;  // tid = tp
    int c = classes[n];
    const float4* Xv = (const float4*)(X + (size_t)n * T_ * D_);
    float4* Xlv = (float4*)Xl;
    Xlv[tid]       = Xv[tid];
    Xlv[tid + 256] = Xv[tid + 256];
    __syncthreads();
    const float* P = pi + (size_t)c * T_ * TP_ + tid;   // column tp = tid
    float acc[D_];
#pragma unroll
    for (int d = 0; d < D_; ++d) acc[d] = 0.f;
    for (int t = 0; t < T_; ++t) {
        float p = P[t * TP_];               // coalesced across threads
        if (p != 0.f) {                     // monotone path: ~1.5 hits/col
#pragma unroll
            for (int d = 0; d < D_; ++d) acc[d] += p * Xl[t * D_ + d];
        }
    }
    float4* O = (float4*)(XW + (size_t)n * K_ + tid * D_);
    O[0] = make_float4(acc[0], acc[1], acc[2], acc[3]);
    O[1] = make_float4(acc[4], acc[5], acc[6], acc[7]);
}

// ---------------- Kernel 5: WMMA f32 GEMM + fused epilogue -----------------
// out[n,j] = C1n[n] + C2[classes[n],j] - 2 * dot(XW[n,:], Yf[j,:]),  Kdim=2048
// Block tile 128x64 (8 waves, 4x2), wave tile 32x32 (2x2 register blocking).
#define BM 128
#define BN 64
#define KB 64
#define LSTR (KB + 4)   // 68: float4-aligned LDS stores, conflict-free reads

__global__ __launch_bounds__(256)
void k_gemm(const float* __restrict__ XW, const float* __restrict__ Yf,
            const float* __restrict__ C1n, const float* __restrict__ C2,
            const int* __restrict__ classes, float* __restrict__ out) {
    __shared__ float As[BM * LSTR];   // 34816 B
    __shared__ float Bs[BN * LSTR];   // 17408 B
    int tid  = threadIdx.x;
    int wave = tid >> 5, lane = tid & 31;
    int half = lane >> 4, l16 = lane & 15;
    int nBase = blockIdx.y * BM;
    int jBase = blockIdx.x * BN;
    int wr = wave >> 1, wc = wave & 1;      // 4x2 waves -> 128x64 tile

    v8f acc00 = {}, acc01 = {}, acc10 = {}, acc11 = {};

    for (int k0 = 0; k0 < K_; k0 += KB) {
        // Stage A tile: 128 rows x 64 k (2048 float4, 8 per thread, coalesced)
#pragma unroll
        for (int i = 0; i < 8; ++i) {
            int idx = tid + i * 256;
            int r = idx >> 4, q = idx & 15;
            float4 v = *(const float4*)(XW + (size_t)(nBase + r) * K_ + k0 + q * 4);
            *(float4*)(As + r * LSTR + q * 4) = v;
        }
        // Stage B tile: 64 rows x 64 k (1024 float4, 4 per thread)
#pragma unroll
        for (int i = 0; i < 4; ++i) {
            int idx = tid + i * 256;
            int r = idx >> 4, q = idx & 15;
            float4 v = *(const float4*)(Yf + (size_t)(jBase + r) * K_ + k0 + q * 4);
            *(float4*)(Bs + r * LSTR + q * 4) = v;
        }
        __syncthreads();

        // WMMA f32 16x16x4: A/B lane layout = {row l16, k = 2*half + 0/1}
        const float* Ab = As + (wr * 32 + l16) * LSTR + 2 * half;
        const float* Bb = Bs + (wc * 32 + l16) * LSTR + 2 * half;
#pragma unroll
        for (int kk = 0; kk < KB; kk += 4) {
            v2f a0 = *(const v2f*)(Ab + kk);
            v2f a1 = *(const v2f*)(Ab + 16 * LSTR + kk);
            v2f b0 = *(const v2f*)(Bb + kk);
            v2f b1 = *(const v2f*)(Bb + 16 * LSTR + kk);
            // 4 WMMAs reuse 2 A-frags and 2 B-frags; independent accumulators
            acc00 = __builtin_amdgcn_wmma_f32_16x16x4_f32(
                false, a0, false, b0, (short)0, acc00, false, false);
            acc01 = __builtin_amdgcn_wmma_f32_16x16x4_f32(
                false, a0, false, b1, (short)0, acc01, false, false);
            acc10 = __builtin_amdgcn_wmma_f32_16x16x4_f32(
                false, a1, false, b0, (short)0, acc10, false, false);
            acc11 = __builtin_amdgcn_wmma_f32_16x16x4_f32(
                false, a1, false, b1, (short)0, acc11, false, false);
        }
        __syncthreads();
    }

    // Epilogue: C/D layout -> VGPR v holds M = v + 8*half, N = l16
    int rowBase = nBase + wr * 32;
    int colBase = jBase + wc * 32;
#pragma unroll
    for (int sm = 0; sm < 2; ++sm) {
#pragma unroll
        for (int sn = 0; sn < 2; ++sn) {
            v8f acc = (sm == 0) ? (sn == 0 ? acc00 : acc01)
                                : (sn == 0 ? acc10 : acc11);
            int col = colBase + sn * 16 + l16;
#pragma unroll
            for (int v = 0; v < 8; ++v) {
                int row = rowBase + sm * 16 + v + 8 * half;
                int cls = classes[row];
                out[(size_t)row * NY_ + col] =
                    C1n[row] + C2[cls * NY_ + col] - 2.0f * acc[v];
            }
        }
    }
}

extern "C" void kernel_launch(void* const* d_in, const int* in_sizes, int n_in,
                              void* d_out, int out_size, void* d_ws, size_t ws_size,
                              hipStream_t stream) {
    const float* X       = (const float*)d_in[0];   // [N, T, D]
    const float* Y       = (const float*)d_in[1];   // [NY, TP, D]
    const float* pi      = (const float*)d_in[2];   // [C, T, TP]
    const int*   classes = (const int*)d_in[3];     // [N]
    float* out = (float*)d_out;                     // [N, NY]

    float* ws     = (float*)d_ws;
    float* XW     = ws;                             // N*K = 2,097,152 floats
    float* rowsum = XW + (size_t)N_ * K_;           // C*T
    float* colsum = rowsum + C_ * T_;               // C*TP
    float* C2     = colsum + C_ * TP_;              // C*NY
    float* C1n    = C2 + C_ * NY_;                  // N

    k_sums<<<C_,  256, 0, stream>>>(pi, rowsum, colsum);
    k_c2  <<<NY_, 256, 0, stream>>>(Y, colsum, C2);
    k_c1  <<<N_,  256, 0, stream>>>(X, rowsum, classes, C1n);
    k_warp<<<N_,  256, 0, stream>>>(X, pi, classes, XW);

    dim3 grid(NY_ / BN, N_ / BM);                   // (16, 8)
    k_gemm<<<grid, 256, 0, stream>>>(XW, Y, C1n, C2, classes, out);
}